// Model_36850819399702
// MI455X (gfx1250) — compile-verified
//
#include <hip/hip_runtime.h>
#include <hip/hip_bf16.h>

#define DIMS   10000
#define NLEV   100
#define SZ     617
#define NCLS   26
#define BATCH  32

// ---------------- Encode kernel ----------------
#define TILE_D 256
#define BLK_A  128
#define DPT    2     // dims per thread (float2)

#if defined(__has_builtin)
#if __has_builtin(__builtin_amdgcn_global_load_async_to_lds_b128)
#define HAVE_ASYNC_LDS 1
#endif
#endif

#ifdef HAVE_ASYNC_LDS
typedef int v4i_ __attribute__((vector_size(16)));
typedef __attribute__((address_space(1))) v4i_ as1_v4i;
typedef __attribute__((address_space(3))) v4i_ as3_v4i;

__device__ __forceinline__ as1_v4i* to_glob(const void* p) {
    return (as1_v4i*)(unsigned long long)p;
}
__device__ __forceinline__ as3_v4i* to_lds(void* p) {
    // Generic LDS addresses carry the LDS byte offset in bits [31:0].
    return (as3_v4i*)(unsigned int)(unsigned long long)p;
}
#endif

__global__ __launch_bounds__(BLK_A) void encode_kernel(
    const float* __restrict__ x,
    const float* __restrict__ idw,
    const float* __restrict__ lvl,
    float* __restrict__ enc)
{
    __shared__ float         lvl_lds[NLEV * TILE_D];   // 100*256*4 = 102400 B
    __shared__ unsigned char idx_lds[SZ * BATCH];      // 617*32    =  19744 B

    const int tid   = threadIdx.x;
    const int dbase = blockIdx.x * TILE_D;
    const bool full_tile = (dbase + TILE_D) <= DIMS;

#ifdef HAVE_ASYNC_LDS
    if (full_tile) {
        // Async DMA the level-table slice straight into LDS (no VGPR bounce).
        // 100 rows * 256 floats = 6400 x 16B chunks, 64 chunks per row.
        for (int u = tid; u < NLEV * (TILE_D / 4); u += BLK_A) {
            const int l   = u >> 6;          // u / 64
            const int j16 = u & 63;          // chunk within row
            const float* gsrc = lvl + (size_t)l * DIMS + dbase + j16 * 4;
            float*       ldst = &lvl_lds[l * TILE_D + j16 * 4];
            __builtin_amdgcn_global_load_async_to_lds_b128(
                to_glob(gsrc), to_lds(ldst), /*offset=*/0, /*cpol=*/0);
        }
    } else
#endif
    {
        for (int t = tid; t < NLEV * TILE_D; t += BLK_A) {
            const int l = t / TILE_D;
            const int j = t - l * TILE_D;
            const int d = dbase + j;
            lvl_lds[t] = (d < DIMS) ? lvl[(size_t)l * DIMS + d] : 0.0f;
        }
    }

    // Overlapped with the async copies: quantize x -> level indices
    // (round-half-even matches jnp.round), s-major bytes.
    for (int t = tid; t < SZ * BATCH; t += BLK_A) {
        const int s = t >> 5;
        const int b = t & 31;
        float r = __builtin_rintf(x[b * SZ + s] * (float)(NLEV - 1));
        int   l = (int)r;
        l = l < 0 ? 0 : (l > NLEV - 1 ? NLEV - 1 : l);
        idx_lds[t] = (unsigned char)l;
    }

#ifdef HAVE_ASYNC_LDS
#if __has_builtin(__builtin_amdgcn_s_wait_asynccnt)
    __builtin_amdgcn_s_wait_asynccnt(0);
#else
    asm volatile("s_wait_asynccnt 0" ::: "memory");
#endif
#endif
    __syncthreads();

    const int d0 = dbase + tid * DPT;
    if (d0 + 1 < DIMS) {
        float acc[BATCH][DPT];
#pragma unroll
        for (int b = 0; b < BATCH; ++b) { acc[b][0] = 0.0f; acc[b][1] = 0.0f; }

        const int lbase = tid * DPT;
        for (int s = 0; s < SZ; ++s) {
            if (s + 8 < SZ)
                __builtin_prefetch(idw + (size_t)(s + 8) * DIMS + d0, 0, 0);
            const float2 idv = *(const float2*)(idw + (size_t)s * DIMS + d0);
            const unsigned int* iw =
                (const unsigned int*)&idx_lds[s * BATCH];   // 8 dwords, 4B aligned
#pragma unroll
            for (int w = 0; w < 8; ++w) {
                const unsigned int packed = iw[w];
#pragma unroll
                for (int j = 0; j < 4; ++j) {
                    const int b = w * 4 + j;
                    const int l = (packed >> (8 * j)) & 0xFF;
                    const float2 lv = *(const float2*)&lvl_lds[l * TILE_D + lbase];
                    acc[b][0] = fmaf(idv.x, lv.x, acc[b][0]);
                    acc[b][1] = fmaf(idv.y, lv.y, acc[b][1]);
                }
            }
        }
        // hard_quantize -> ws
#pragma unroll
        for (int b = 0; b < BATCH; ++b) {
            enc[(size_t)b * DIMS + d0]     = acc[b][0] > 0.0f ? 1.0f : -1.0f;
            enc[(size_t)b * DIMS + d0 + 1] = acc[b][1] > 0.0f ? 1.0f : -1.0f;
        }
    }
}

// ---------------- WMMA classifier GEMM ----------------
// logit[b,c] = sum_d enc[b,d] * W[c,d], exact f32 via V_WMMA_F32_16X16X4_F32.
typedef __attribute__((ext_vector_type(2))) float v2f;
typedef __attribute__((ext_vector_type(8))) float v8f;

#define KSPLIT 10
#define KSPAN  (DIMS / KSPLIT)   // 1000 dims per split, multiple of 4

__global__ __launch_bounds__(32) void gemm_wmma_kernel(
    const float* __restrict__ enc,
    const float* __restrict__ W,
    float* __restrict__ partial)   // [KSPLIT][32][32]
{
    const int lane = threadIdx.x;
    const int mt = blockIdx.x;            // 0..1  (batch tiles)
    const int ct = blockIdx.y;            // 0..1  (class tiles)
    const int ks = blockIdx.z;            // 0..KSPLIT-1

    const int mn  = lane & 15;            // row/col within tile
    const int kb  = (lane >> 4) * 2;      // K pair base for this lane half

    const int m = mt * 16 + mn;           // 0..31, always valid
    const int n = ct * 16 + mn;           // class index, may be >= 26
    const bool nvalid = (n < NCLS);
    const int nc = nvalid ? n : 0;        // clamp; zero via cndmask below

    const float* arow = enc + (size_t)m  * DIMS;
    const float* brow = W   + (size_t)nc * DIMS;

    v8f c = {};
    const int dstart = ks * KSPAN;
    const int dend   = dstart + KSPAN;
    for (int d = dstart; d < dend; d += 4) {
        v2f a = *(const v2f*)(arow + d + kb);
        v2f b = *(const v2f*)(brow + d + kb);
        b = nvalid ? b : (v2f)(0.0f);     // select, keeps EXEC all-ones at WMMA
        c = __builtin_amdgcn_wmma_f32_16x16x4_f32(
                /*neg_a=*/false, a, /*neg_b=*/false, b,
                /*c_mod=*/(short)0, c, /*reuse_a=*/false, /*reuse_b=*/false);
    }

    // C/D layout: VGPR v, lanes 0-15 -> M=v, lanes 16-31 -> M=v+8, N=lane%16.
    const int half = lane >> 4;
#pragma unroll
    for (int v = 0; v < 8; ++v) {
        const int row = mt * 16 + v + 8 * half;
        const int col = ct * 16 + mn;
        partial[(size_t)ks * (32 * 32) + row * 32 + col] = c[v];
    }
}

// ---------------- Split-K reduction ----------------
__global__ void reduce_kernel(const float* __restrict__ partial,
                              float* __restrict__ out)
{
    const int t = blockIdx.x * blockDim.x + threadIdx.x;
    if (t >= BATCH * NCLS) return;
    const int b = t / NCLS;
    const int cc = t - b * NCLS;
    float s = 0.0f;
#pragma unroll
    for (int k = 0; k < KSPLIT; ++k)
        s += partial[(size_t)k * (32 * 32) + b * 32 + cc];
    out[t] = s;
}

extern "C" void kernel_launch(void* const* d_in, const int* in_sizes, int n_in,
                              void* d_out, int out_size, void* d_ws, size_t ws_size,
                              hipStream_t stream) {
    const float* x   = (const float*)d_in[0];   // (32, 617)
    const float* idw = (const float*)d_in[1];   // (617, 10000)
    const float* lvl = (const float*)d_in[2];   // (100, 10000)
    const float* W   = (const float*)d_in[3];   // (26, 10000)
    float* out = (float*)d_out;                 // (32, 26)

    float* enc     = (float*)d_ws;               // 32*10000 floats = 1.28 MB
    float* partial = enc + (size_t)BATCH * DIMS; // KSPLIT*32*32 floats = 40 KB

    dim3 gA((DIMS + TILE_D - 1) / TILE_D);      // 40 blocks
    encode_kernel<<<gA, BLK_A, 0, stream>>>(x, idw, lvl, enc);

    dim3 gB(2, 2, KSPLIT);                      // 40 waves, one per 16x16 K-slice
    gemm_wmma_kernel<<<gB, 32, 0, stream>>>(enc, W, partial);

    reduce_kernel<<<(BATCH * NCLS + 255) / 256, 256, 0, stream>>>(partial, out);
}